// AdaptiveUpSample_9938554323451
// MI455X (gfx1250) — compile-verified
//
#include <hip/hip_runtime.h>
#include <hip/hip_bf16.h>

// ---------------------------------------------------------------------------
// AdaptiveUpSample for MI455X (gfx1250, wave32)
//   Stage 1: 1x1-conv offset GEMM via V_WMMA_F32_16X16X4_F32 (exact f32 WMMA).
//            Each wave: 32(M pixels) x 32(N chan) x 256(K), 4 accumulators,
//            B (weight) operands reused across both M-tiles -> 8 loads / 4 WMMA.
//            Fused epilogue emits clamped absolute sample coords into d_ws.
//   Stage 2: bilinear border-clamped gather; geometry mapped onto the launch
//            grid so it is division-free; fully coalesced 210 MB store stream.
// Bandwidth-bound problem (~300 MB moved, ~1.3 GFLOP at 23.3 TB/s ~ 13 us).
// ---------------------------------------------------------------------------

typedef float v2f __attribute__((ext_vector_type(2)));
typedef float v8f __attribute__((ext_vector_type(8)));

#define B_      8
#define C_      256
#define H_      80
#define W_      80
#define HW      (H_ * W_)          // 6400
#define GROUPS  4
#define SH      (2 * H_)           // 160
#define SW      (2 * W_)           // 160
#define NPIX    (B_ * HW)          // 51200 GEMM rows (pixels)
#define NT2     (NPIX / 32)        // 1600 double-tiles (one per wave)

// ---------------------------------------------------------------------------
// Kernel 1: per-wave 32x32x256 GEMM using f32 WMMA.
// coords layout: coords[((b*16 + j)*HW + h*W + w)*2 + {0:ix, 1:iy}]
// ---------------------------------------------------------------------------
__global__ __launch_bounds__(256)
void adaptive_up_offsets_wmma(const float* __restrict__ x,
                              const float* __restrict__ w_off,
                              const float* __restrict__ b_off,
                              float* __restrict__ coords) {
    const int lane  = threadIdx.x & 31;
    const int wid   = threadIdx.x >> 5;
    const int tile2 = blockIdx.x * 8 + wid;          // wave-uniform
    if (tile2 >= NT2) return;                        // uniform: EXEC stays all-1

    const int nIdx = lane & 15;                      // A: pixel row | B: N column
    const int kGrp = lane >> 4;                      // 0 -> K{0,1}, 1 -> K{2,3}
    const int p0   = tile2 * 32;                     // first pixel (32-aligned)

    // 32 | HW, so the 32-pixel double-tile never straddles a batch.
    const int bb = p0 / HW;                          // scalar
    const int q0 = p0 - bb * HW;                     // scalar, h*W + w

    // A operands: lane holds pixel q0(+16) + nIdx at K = k0 + 2*kGrp + {0,1}
    const float* xa = x + (size_t)bb * (C_ * HW) + q0 + nIdx;   // + k*HW per step

    // B operands: lane holds N = nIdx (x-offsets) / nIdx+16 (y-offsets)
    const float* wb0 = w_off + (size_t)nIdx * C_;
    const float* wb1 = w_off + (size_t)(nIdx + 16) * C_;

    v8f acc00 = {}, acc01 = {};   // M-tile 0: off_x / off_y
    v8f acc10 = {}, acc11 = {};   // M-tile 1: off_x / off_y

    for (int k0 = 0; k0 < C_; k0 += 4) {
        const int k = k0 + 2 * kGrp;
        const float* xk = xa + (size_t)k * HW;
        v2f aF0, aF1, bF0, bF1;
        aF0.x = xk[0];       aF0.y = xk[HW];          // M-tile 0, K, K+1
        aF1.x = xk[16];      aF1.y = xk[HW + 16];     // M-tile 1, K, K+1
        bF0.x = wb0[k];      bF0.y = wb0[k + 1];
        bF1.x = wb1[k];      bF1.y = wb1[k + 1];
        // 8-arg form: (neg_a, A, neg_b, B, c_mod, C, reuse_a, reuse_b)
        acc00 = __builtin_amdgcn_wmma_f32_16x16x4_f32(false, aF0, false, bF0,
                                                      (short)0, acc00, false, false);
        acc01 = __builtin_amdgcn_wmma_f32_16x16x4_f32(false, aF0, false, bF1,
                                                      (short)0, acc01, false, false);
        acc10 = __builtin_amdgcn_wmma_f32_16x16x4_f32(false, aF1, false, bF0,
                                                      (short)0, acc10, false, false);
        acc11 = __builtin_amdgcn_wmma_f32_16x16x4_f32(false, aF1, false, bF1,
                                                      (short)0, acc11, false, false);
    }

    // C/D layout: VGPR r, lane l -> (M = r + 8*(l>=16), N = l&15).
    const int j = nIdx;                        // offset channel j = grp*4+a*2+b2
    const float bx = b_off[j];
    const float by = b_off[j + 16];
    const float initx = (j & 1)        ? 0.25f : -0.25f;   // h[b2]
    const float inity = ((j >> 1) & 1) ? 0.25f : -0.25f;   // h[a]

    #pragma unroll
    for (int st = 0; st < 2; ++st) {
        const v8f accx = st ? acc10 : acc00;
        const v8f accy = st ? acc11 : acc01;
        // Each 16-pixel sub-tile stays inside one row: q0s%80 in {0,16,32,48,64}.
        const int q0s = q0 + 16 * st;                 // scalar
        const int h   = q0s / W_;                     // scalar
        const int w0  = q0s - h * W_;                 // scalar
        const size_t cbase = ((size_t)(bb * 16 + j) * HW + q0s) * 2;
        #pragma unroll
        for (int r = 0; r < 8; ++r) {
            const int d = r + 8 * kGrp;               // pixel within sub-tile
            const float offx = fmaf(accx[r] + bx, 0.25f, initx);
            const float offy = fmaf(accy[r] + by, 0.25f, inity);
            const float ix = fminf(fmaxf((float)(w0 + d) + offx, 0.0f), (float)(W_ - 1));
            const float iy = fminf(fmaxf((float)h        + offy, 0.0f), (float)(H_ - 1));
            coords[cbase + 2 * d]     = ix;
            coords[cbase + 2 * d + 1] = iy;
        }
    }
}

// ---------------------------------------------------------------------------
// Kernel 2: geometry carried by the launch grid (division-free).
//   block = (SW, 2)  -> xo = threadIdx.x, 2 output rows per block (10 waves)
//   grid  = (SH/2, B_*GROUPS)
// Each thread bilinearly samples the 64 channels of its group.
// ---------------------------------------------------------------------------
__global__ __launch_bounds__(SW * 2)
void adaptive_up_gather(const float* __restrict__ x,
                        const float* __restrict__ coords,
                        float* __restrict__ out) {
    const int xo  = threadIdx.x;
    const int yo  = blockIdx.x * 2 + threadIdx.y;
    const int bg  = blockIdx.y;
    const int grp = bg & (GROUPS - 1);
    const int b   = bg >> 2;

    const int h  = yo >> 1, a  = yo & 1;
    const int w  = xo >> 1, b2 = xo & 1;
    const int j  = grp * 4 + a * 2 + b2;
    const int q  = h * W_ + w;

    const size_t ci = ((size_t)(b * 16 + j) * HW + q) * 2;
    const float ix = coords[ci];
    const float iy = coords[ci + 1];

    const float xf = floorf(ix), yf = floorf(iy);
    const int   x0 = (int)xf,    y0 = (int)yf;
    const float wx = ix - xf,    wy = iy - yf;
    const int   x1 = min(x0 + 1, W_ - 1);
    const int   y1 = min(y0 + 1, H_ - 1);

    const float w00 = (1.0f - wy) * (1.0f - wx);
    const float w01 = (1.0f - wy) * wx;
    const float w10 = wy * (1.0f - wx);
    const float w11 = wy * wx;

    const int i00 = y0 * W_ + x0, i01 = y0 * W_ + x1;
    const int i10 = y1 * W_ + x0, i11 = y1 * W_ + x1;

    const float* xb = x   + (size_t)(b * C_ + grp * 64) * HW;
    float*       ob = out + ((size_t)(b * C_ + grp * 64) * SH + yo) * SW + xo;

    #pragma unroll 4
    for (int ch = 0; ch < 64; ++ch) {
        const float* xc = xb + (size_t)ch * HW;
        const float v = xc[i00] * w00 + xc[i01] * w01 +
                        xc[i10] * w10 + xc[i11] * w11;
        ob[(size_t)ch * (SH * SW)] = v;
    }
}

// ---------------------------------------------------------------------------
extern "C" void kernel_launch(void* const* d_in, const int* in_sizes, int n_in,
                              void* d_out, int out_size, void* d_ws, size_t ws_size,
                              hipStream_t stream) {
    const float* x     = (const float*)d_in[0];   // (8,256,80,80)
    const float* w_off = (const float*)d_in[1];   // (32,256,1,1)
    const float* b_off = (const float*)d_in[2];   // (32,)
    float*       out   = (float*)d_out;           // (8,256,160,160)
    float*       crd   = (float*)d_ws;            // (8,16,6400,2) f32 = 6.55 MB

    // Stage 1: 1600 double-tiles, 8 waves per 256-thread block -> 200 blocks.
    adaptive_up_offsets_wmma<<<(NT2 + 7) / 8, 256, 0, stream>>>(x, w_off, b_off, crd);

    // Stage 2: grid carries (yo-pair, b*g); block carries (xo, yo within pair).
    adaptive_up_gather<<<dim3(SH / 2, B_ * GROUPS), dim3(SW, 2), 0, stream>>>(x, crd, out);
}